// KSparse_33346126086434
// MI455X (gfx1250) — compile-verified
//
#include <hip/hip_runtime.h>
#include <stdint.h>

#define COLS   8192
#define TPB    256
#define VPT    32          // values per thread = COLS / TPB
#define CHUNKS (COLS / 4)  // 128-bit chunks per row
#define NWAVE  (TPB / 32)

#define AS1 __attribute__((address_space(1)))
#define AS3 __attribute__((address_space(3)))

typedef int   v4i __attribute__((ext_vector_type(4)));
typedef float v4f __attribute__((ext_vector_type(4)));

#if defined(__has_builtin)
#  if __has_builtin(__builtin_amdgcn_global_load_async_to_lds_b128)
#    define USE_ASYNC_LDS 1
#  endif
#  if __has_builtin(__builtin_amdgcn_s_wait_asynccnt)
#    define HAVE_WAIT_ASYNC_BUILTIN 1
#  endif
#endif

// Order-preserving float32 -> uint32 key (larger float => larger key).
__device__ __forceinline__ unsigned f2key(float f) {
    unsigned u = __float_as_uint(f);
    unsigned s = (unsigned)((int)u >> 31);        // all-ones if negative
    return u ^ (s | 0x80000000u);
}
// Inverse transform.
__device__ __forceinline__ float key2f(unsigned k) {
    unsigned s = (unsigned)((int)k >> 31);        // all-ones if key MSB set (orig positive)
    return __uint_as_float(k ^ (~s | 0x80000000u));
}

__global__ __launch_bounds__(TPB) void
topk_row_mask_kernel(const float* __restrict__ x,
                     const int*   __restrict__ kp,
                     float*       __restrict__ out)
{
    __shared__ v4f      s_data4[CHUNKS];            // 32 KB row staging
    __shared__ unsigned s_whist[NWAVE * 256];       // wave-private histograms (8 KB)
    __shared__ unsigned s_comb[256];                // combined histogram
    __shared__ unsigned s_sel[2];                   // {selected bin, new k_rem}

    const int  tid  = threadIdx.x;
    const int  wid  = tid >> 5;                     // wave32
    const long row  = blockIdx.x;
    const float* rowp = x + row * (long)COLS;

    // ---- Row -> LDS via gfx1250 async global->LDS path (ASYNCcnt) ----
#if defined(USE_ASYNC_LDS)
#pragma unroll
    for (int j = 0; j < 8; ++j) {
        const int chunk = j * TPB + tid;
        __builtin_amdgcn_global_load_async_to_lds_b128(
            (AS1 v4i*)(rowp + chunk * 4),
            (AS3 v4i*)&s_data4[chunk],
            0, 0);
    }
#  if defined(HAVE_WAIT_ASYNC_BUILTIN)
    __builtin_amdgcn_s_wait_asynccnt(0);
#  else
    asm volatile("s_wait_asynccnt 0" ::: "memory");
#  endif
#else
    {
        const v4f* g4 = (const v4f*)rowp;
#pragma unroll
        for (int j = 0; j < 8; ++j) {
            const int chunk = j * TPB + tid;
            s_data4[chunk] = g4[chunk];
        }
    }
#endif
    __syncthreads();

    // ---- Pull this thread's 32 values into VGPRs as sortable keys ----
    unsigned keys[VPT];
#pragma unroll
    for (int j = 0; j < 8; ++j) {
        v4f v = s_data4[j * TPB + tid];
        keys[4 * j + 0] = f2key(v.x);
        keys[4 * j + 1] = f2key(v.y);
        keys[4 * j + 2] = f2key(v.z);
        keys[4 * j + 3] = f2key(v.w);
    }

    unsigned k_rem  = (unsigned)kp[0];   // uniform scalar load
    unsigned prefix = 0u;
    unsigned pmask  = 0u;

    // ---- 4-pass radix select for the k-th largest key ----
    for (int shift = 24; shift >= 0; shift -= 8) {
        // zero wave-private histograms
#pragma unroll
        for (int w = 0; w < NWAVE; ++w) s_whist[w * 256 + tid] = 0u;
        __syncthreads();

        unsigned* myh = &s_whist[wid * 256];
#pragma unroll
        for (int i = 0; i < VPT; ++i) {
            const unsigned kk = keys[i];
            if ((kk & pmask) == prefix)
                atomicAdd(&myh[(kk >> shift) & 0xFFu], 1u);
        }
        __syncthreads();

        // combine the 8 private copies: thread t owns bin t
        unsigned tot = 0u;
#pragma unroll
        for (int w = 0; w < NWAVE; ++w) tot += s_whist[w * 256 + tid];
        s_comb[tid] = tot;
        __syncthreads();

        // wave 0: suffix-select the bin containing the k_rem-th largest
        if (tid < 32) {
            unsigned cnt[8];
            unsigned g = 0u;
#pragma unroll
            for (int i = 0; i < 8; ++i) { cnt[i] = s_comb[tid * 8 + i]; g += cnt[i]; }

            // inclusive suffix sum across lanes: S[l] = sum_{m>=l} g[m]
            unsigned S = g;
#pragma unroll
            for (int off = 1; off < 32; off <<= 1) {
                unsigned v = __shfl_down(S, (unsigned)off, 32);
                if (tid + off < 32) S += v;
            }
            const unsigned above = S - g;    // count in bins >= (tid+1)*8
            if (S >= k_rem && above < k_rem) {
                unsigned cum = above;
#pragma unroll
                for (int i = 7; i >= 0; --i) {
                    const unsigned nc = cum + cnt[i];
                    if (nc >= k_rem && cum < k_rem) {
                        s_sel[0] = (unsigned)(tid * 8 + i);
                        s_sel[1] = k_rem - cum;
                    }
                    cum = nc;
                }
            }
        }
        __syncthreads();

        prefix |= s_sel[0] << shift;
        pmask  |= 0xFFu << shift;
        k_rem   = s_sel[1];
        // no extra barrier needed: s_sel is next written only after 3 barriers
    }

    const unsigned kth = prefix;   // exact bit pattern of the k-th largest value

    // ---- Masked write-back, nontemporal (result is never re-read) ----
    float* orow = out + row * (long)COLS;
#pragma unroll
    for (int j = 0; j < 8; ++j) {
        const int chunk = j * TPB + tid;
        v4f o;
        o.x = (keys[4 * j + 0] >= kth) ? key2f(keys[4 * j + 0]) : 0.0f;
        o.y = (keys[4 * j + 1] >= kth) ? key2f(keys[4 * j + 1]) : 0.0f;
        o.z = (keys[4 * j + 2] >= kth) ? key2f(keys[4 * j + 2]) : 0.0f;
        o.w = (keys[4 * j + 3] >= kth) ? key2f(keys[4 * j + 3]) : 0.0f;
        __builtin_nontemporal_store(o, (v4f*)orow + chunk);
    }
}

extern "C" void kernel_launch(void* const* d_in, const int* in_sizes, int n_in,
                              void* d_out, int out_size, void* d_ws, size_t ws_size,
                              hipStream_t stream) {
    const float* x  = (const float*)d_in[0];
    const int*   kp = (const int*)d_in[1];
    float*       o  = (float*)d_out;

    const int rows = in_sizes[0] / COLS;   // 8192
    if (rows <= 0) return;

    dim3 grid((unsigned)rows), block(TPB);
    topk_row_mask_kernel<<<grid, block, 0, stream>>>(x, kp, o);
}